// HeteroGraphUNet_50620484551195
// MI455X (gfx1250) — compile-verified
//
#include <hip/hip_runtime.h>
#include <cmath>

#define HH 64
#define NHEADS 4

typedef float v2f __attribute__((ext_vector_type(2)));
typedef float v8f __attribute__((ext_vector_type(8)));

// ---------------- device helpers ----------------
__device__ __forceinline__ unsigned f2ord(float f) {
  unsigned u = __float_as_uint(f);
  return (u & 0x80000000u) ? ~u : (u | 0x80000000u);
}
__device__ __forceinline__ float ord2f(unsigned u) {
  return __uint_as_float((u & 0x80000000u) ? (u & 0x7FFFFFFFu) : ~u);
}
__device__ __forceinline__ float geluf(float x) {
  return 0.5f * x * (1.0f + erff(x * 0.70710678118654752440f));
}
__device__ __forceinline__ float leakyf(float x) { return x > 0.f ? x : 0.2f * x; }
__device__ __forceinline__ float wredf(float v) {
#pragma unroll
  for (int o = 16; o > 0; o >>= 1) v += __shfl_xor(v, o, 32);
  return v;
}

// ---------------- GEMM via WMMA f32 16x16x4 ----------------
// Y[M,N] = X[M,K] @ W[K,N] (+bias) (+=Y if acc). One wave per 16x16 tile.
// A layout: lanes 0-15 hold M=l, VGPR0:K=2*half, VGPR1:K=2*half+1
// B layout (symmetric): lanes hold N=l, VGPR0:K=2*half, VGPR1:K=2*half+1
// C layout: VGPR r -> row r + 8*half, col l.
__global__ void k_gemm(const float* __restrict__ X, const float* __restrict__ W,
                       const float* __restrict__ B, float* Y,
                       int M, int K, int N, int acc) {
  int lane = threadIdx.x & 31;
  int half = lane >> 4, l = lane & 15;
  int row = blockIdx.x * 16 + l;
  int col = blockIdx.y * 16 + l;
  int rA = row < M ? row : 0;  float fA = row < M ? 1.f : 0.f;
  int cB = col < N ? col : 0;  float fB = col < N ? 1.f : 0.f;
  v8f c = {};
  if (acc) {
#pragma unroll
    for (int r = 0; r < 8; ++r) {
      int orow = blockIdx.x * 16 + r + 8 * half;
      c[r] = (orow < M && col < N) ? Y[(size_t)orow * N + col] : 0.f;
    }
  }
  for (int k = 0; k < K; k += 4) {  // all K here are multiples of 4
    int ka = k + 2 * half;
    v2f a, b;
    a.x = X[(size_t)rA * K + ka] * fA;
    a.y = X[(size_t)rA * K + ka + 1] * fA;
    b.x = W[(size_t)ka * N + cB] * fB;
    b.y = W[(size_t)(ka + 1) * N + cB] * fB;
    c = __builtin_amdgcn_wmma_f32_16x16x4_f32(false, a, false, b, (short)0, c,
                                              false, false);
  }
#pragma unroll
  for (int r = 0; r < 8; ++r) {
    int orow = blockIdx.x * 16 + r + 8 * half;
    if (orow < M && col < N) {
      float v = c[r];
      if (B) v += B[col];
      Y[(size_t)orow * N + col] = v;
    }
  }
}

// ---------------- LayerNorm (+residual) (+GELU), wave per row ----------------
__global__ void k_ln_act(float* Y, const float* X, const float* R,
                         const float* __restrict__ G, const float* __restrict__ Bt,
                         int M, int D, int act) {
  int wave = (blockIdx.x * blockDim.x + threadIdx.x) >> 5;
  int lane = threadIdx.x & 31;
  if (wave >= M) return;
  int per = D >> 5;  // D = 64 or 128
  float v[4];
  float s = 0.f;
  for (int i = 0; i < per; ++i) {
    int d = lane + 32 * i;
    float t = X[(size_t)wave * D + d];
    if (R) t += R[(size_t)wave * D + d];
    v[i] = t; s += t;
  }
  s = wredf(s);
  float mu = s / (float)D;
  float q = 0.f;
  for (int i = 0; i < per; ++i) { float d0 = v[i] - mu; q += d0 * d0; }
  q = wredf(q);
  float rs = rsqrtf(q / (float)D + 1e-5f);
  for (int i = 0; i < per; ++i) {
    int d = lane + 32 * i;
    float t = (v[i] - mu) * rs * G[d] + Bt[d];
    if (act) t = geluf(t);
    Y[(size_t)wave * D + d] = t;
  }
}

// ---------------- fills / elementwise ----------------
__global__ void k_fill_f32(float* p, float v, int n) {
  int t = blockIdx.x * blockDim.x + threadIdx.x;
  if (t < n) p[t] = v;
}
__global__ void k_fill_u32(unsigned* p, unsigned v, int n) {
  int t = blockIdx.x * blockDim.x + threadIdx.x;
  if (t < n) p[t] = v;
}
__global__ void k_fill_i32(int* p, int v, int n) {
  int t = blockIdx.x * blockDim.x + threadIdx.x;
  if (t < n) p[t] = v;
}
__global__ void k_leaky(float* p, int n) {
  int t = blockIdx.x * blockDim.x + threadIdx.x;
  if (t < n) p[t] = leakyf(p[t]);
}

// ---------------- GAT kernels ----------------
// el/er per (node, head): wave per (node,head) dot over 64 feats.
__global__ void k_head_dots(const float* __restrict__ z, const float* __restrict__ al,
                            const float* __restrict__ ar, float* el, float* er, int n) {
  int gid = (blockIdx.x * blockDim.x + threadIdx.x) >> 5;
  int lane = threadIdx.x & 31;
  int node = gid >> 2, h = gid & 3;
  if (node >= n) return;
  const float* zp = z + (size_t)node * 256 + h * 64;
  float z0 = zp[lane], z1 = zp[lane + 32];
  float sl = wredf(z0 * al[h * 64 + lane] + z1 * al[h * 64 + lane + 32]);
  float sr = wredf(z0 * ar[h * 64 + lane] + z1 * ar[h * 64 + lane + 32]);
  if (lane == 0) { el[node * 4 + h] = sl; er[node * 4 + h] = sr; }
}

__global__ void k_edge_logits(const float* __restrict__ el, const float* __restrict__ er,
                              const int* __restrict__ src, const int* __restrict__ dst,
                              const int* __restrict__ mask, float* eb,
                              unsigned* menc, int E) {
  int t = blockIdx.x * blockDim.x + threadIdx.x;
  if (t >= E * NHEADS) return;
  int e = t >> 2, h = t & 3;
  if (mask && !mask[e]) return;
  int s = src[e], d = dst[e];
  float v = leakyf(el[s * 4 + h] + er[d * 4 + h]);
  eb[t] = v;
  atomicMax(&menc[d * 4 + h], f2ord(v));
}

__global__ void k_edge_expsum(float* eb, const unsigned* __restrict__ menc,
                              float* sb, const int* __restrict__ dst,
                              const int* __restrict__ mask, int E) {
  int t = blockIdx.x * blockDim.x + threadIdx.x;
  if (t >= E * NHEADS) return;
  int e = t >> 2, h = t & 3;
  if (mask && !mask[e]) return;
  int d = dst[e];
  float w = expf(eb[t] - ord2f(menc[d * 4 + h]));
  eb[t] = w;
  atomicAdd(&sb[d * 4 + h], w);
}

// weighted scatter: wave per edge, 256 floats (4 heads x 64).
__global__ void k_gat_scatter(const float* __restrict__ z, const float* __restrict__ w,
                              const float* __restrict__ sb, const int* __restrict__ src,
                              const int* __restrict__ dst, const int* __restrict__ mask,
                              float* out, int E) {
  int e = (blockIdx.x * blockDim.x + threadIdx.x) >> 5;
  int lane = threadIdx.x & 31;
  if (e >= E) return;
  if (mask && !mask[e]) return;
  int s = src[e], d = dst[e];
  float a = 0.f;
  if (lane < 4) a = w[(size_t)e * 4 + lane] / (sb[(size_t)d * 4 + lane] + 1e-9f);
  float av[4] = { __shfl(a, 0, 32), __shfl(a, 1, 32), __shfl(a, 2, 32), __shfl(a, 3, 32) };
  const float* zs = z + (size_t)s * 256;
  float* od = out + (size_t)d * 256;
#pragma unroll
  for (int i = 0; i < 8; ++i) {
    int idx = lane + 32 * i;
    atomicAdd(&od[idx], zs[idx] * av[i >> 1]);  // head = idx>>6 = i>>1
  }
}

// mean over heads + bias: y[i][d] = mean_h(out[i][h*64+d] + b[h*64+d])
__global__ void k_hmean(const float* __restrict__ o, const float* __restrict__ bias,
                        float* y, int n) {
  int t = blockIdx.x * blockDim.x + threadIdx.x;
  if (t >= n * HH) return;
  int i = t >> 6, d = t & 63;
  float s = 0.f;
#pragma unroll
  for (int h = 0; h < 4; ++h) s += o[(size_t)i * 256 + h * 64 + d] + bias[h * 64 + d];
  y[t] = s * 0.25f;
}

// ---------------- pooling (radix-bisect top-k) ----------------
__global__ void k_score_enc(const float* __restrict__ s4, float* sc, unsigned* en, int n) {
  int t = blockIdx.x * blockDim.x + threadIdx.x;
  if (t >= n) return;
  float m = (s4[t * 4] + s4[t * 4 + 1] + s4[t * 4 + 2] + s4[t * 4 + 3]) * 0.25f;
  float sig = 1.f / (1.f + expf(-m));
  sc[t] = sig; en[t] = f2ord(sig);
}
// state: [0]=lo [1]=hi [2]=mid [3]=cnt [4]=k
__global__ void k_bisect_init(unsigned* st, unsigned k) {
  st[0] = 0u; st[1] = 0xFFFFFFFFu;
  unsigned d = 0xFFFFFFFFu;
  st[2] = (d >> 1) + (d & 1u); st[3] = 0u; st[4] = k;
}
__global__ void k_count_ge(const unsigned* __restrict__ en, int n, unsigned* st) {
  int t = blockIdx.x * blockDim.x + threadIdx.x;
  if (t < n && en[t] >= st[2]) atomicAdd(&st[3], 1u);
}
__global__ void k_bisect_step(unsigned* st) {
  unsigned lo = st[0], hi = st[1], mid = st[2], cnt = st[3], k = st[4];
  if (cnt >= k) lo = mid; else hi = mid - 1u;
  unsigned d = hi - lo;
  st[0] = lo; st[1] = hi; st[2] = lo + (d >> 1) + (d & 1u); st[3] = 0u;
}
__global__ void k_sel_gt(const unsigned* __restrict__ en, int n, unsigned* st, int* idx, int k) {
  int t = blockIdx.x * blockDim.x + threadIdx.x;
  if (t < n && en[t] > st[0]) {
    unsigned p = atomicAdd(&st[3], 1u);
    if (p < (unsigned)k) idx[p] = t;
  }
}
__global__ void k_sel_eq(const unsigned* __restrict__ en, int n, unsigned* st, int* idx, int k) {
  int t = blockIdx.x * blockDim.x + threadIdx.x;
  if (t < n && en[t] == st[0]) {
    unsigned p = atomicAdd(&st[3], 1u);
    if (p < (unsigned)k) idx[p] = t;
  }
}
__global__ void k_build_map(const int* __restrict__ idx, int* mp, int k) {
  int t = blockIdx.x * blockDim.x + threadIdx.x;
  if (t < k) mp[idx[t]] = t;
}
__global__ void k_gather_scale(const float* __restrict__ x, const int* __restrict__ idx,
                               const float* __restrict__ sc, float* nx, int k) {
  int t = blockIdx.x * blockDim.x + threadIdx.x;
  if (t >= k * HH) return;
  int j = t >> 6, d = t & 63;
  int i0 = idx[j];
  nx[t] = x[(size_t)i0 * HH + d] * (1.f + sc[i0]);
}
__global__ void k_remap(const int* __restrict__ s, const int* __restrict__ d,
                        const int* __restrict__ m, const int* __restrict__ mp,
                        int* ns, int* nd, int* nm, int E) {
  int t = blockIdx.x * blockDim.x + threadIdx.x;
  if (t >= E) return;
  int valid = m ? m[t] : 1;
  int a = valid ? mp[s[t]] : -1;
  int b = valid ? mp[d[t]] : -1;
  int ok = (valid && a >= 0 && b >= 0) ? 1 : 0;
  ns[t] = ok ? a : 0; nd[t] = ok ? b : 0; nm[t] = ok;
}

// ---------------- GCN / SAGE ----------------
__global__ void k_degree(const int* __restrict__ s, const int* __restrict__ d,
                         const int* __restrict__ m, float* dego, float* degi, int E) {
  int t = blockIdx.x * blockDim.x + threadIdx.x;
  if (t >= E) return;
  if (m && !m[t]) return;
  atomicAdd(&dego[s[t]], 1.f);
  atomicAdd(&degi[d[t]], 1.f);
}
__global__ void k_gcn_scatter(const float* __restrict__ xw, const int* __restrict__ s,
                              const int* __restrict__ d, const int* __restrict__ m,
                              const float* __restrict__ dego, float* agg, int E) {
  int e = (blockIdx.x * blockDim.x + threadIdx.x) >> 5;
  int lane = threadIdx.x & 31;
  if (e >= E) return;
  if (m && !m[e]) return;
  int a = s[e], b = d[e];
  float coef = rsqrtf(fmaxf(dego[a], 1.f));
#pragma unroll
  for (int i = 0; i < 2; ++i) {
    int idx = lane + 32 * i;
    atomicAdd(&agg[(size_t)b * HH + idx], xw[(size_t)a * HH + idx] * coef);
  }
}
__global__ void k_gcn_finish(const float* __restrict__ agg, const float* __restrict__ degi,
                             const float* __restrict__ bias, float* y, int n) {
  int t = blockIdx.x * blockDim.x + threadIdx.x;
  if (t >= n * HH) return;
  int i = t >> 6, d = t & 63;
  y[t] = agg[t] * rsqrtf(fmaxf(degi[i], 1.f)) + bias[d];
}
__global__ void k_sage_scatter(const float* __restrict__ x, const int* __restrict__ s,
                               const int* __restrict__ d, const int* __restrict__ m,
                               float* sum, float* cnt, int E) {
  int e = (blockIdx.x * blockDim.x + threadIdx.x) >> 5;
  int lane = threadIdx.x & 31;
  if (e >= E) return;
  if (m && !m[e]) return;
  int a = s[e], b = d[e];
#pragma unroll
  for (int i = 0; i < 4; ++i) {
    int idx = lane + 32 * i;
    atomicAdd(&sum[(size_t)b * 128 + idx], x[(size_t)a * 128 + idx]);
  }
  if (lane == 0) atomicAdd(&cnt[b], 1.f);
}
__global__ void k_sage_div(const float* __restrict__ sum, const float* __restrict__ cnt,
                           float* mean, int n) {
  int t = blockIdx.x * blockDim.x + threadIdx.x;
  if (t >= n * 128) return;
  mean[t] = sum[t] / fmaxf(cnt[t >> 7], 1.f);
}

// ---------------- unpool / concat ----------------
__global__ void k_scatter_rows(const float* __restrict__ x, const int* __restrict__ idx,
                               float* up, int k) {
  int t = blockIdx.x * blockDim.x + threadIdx.x;
  if (t >= k * HH) return;
  int i = t >> 6, c = t & 63;
  up[(size_t)idx[i] * HH + c] = x[t];
}
__global__ void k_copy64(float* dstp, const float* __restrict__ srcp, int n, int doff) {
  int t = blockIdx.x * blockDim.x + threadIdx.x;
  if (t >= n * HH) return;
  int i = t >> 6, c = t & 63;
  dstp[(size_t)i * 128 + doff + c] = srcp[(size_t)i * HH + c];
}

// ---------------- host orchestration ----------------
extern "C" void kernel_launch(void* const* d_in, const int* in_sizes, int n_in,
                              void* d_out, int out_size, void* d_ws, size_t ws_size,
                              hipStream_t stream) {
  const float* feat = (const float*)d_in[0];
  const int* src0 = (const int*)d_in[1];
  const int* dst0 = (const int*)d_in[2];
  const int n0 = in_sizes[0] / 32;
  const int E = in_sizes[1];

  // ---- unpack params (setup_inputs dict insertion order, depth-first) ----
  int pi = 3;
  auto nxt = [&]() { return (const float*)d_in[pi++]; };
  const float* embW = nxt(); const float* embB = nxt();
  const float* embG = nxt(); const float* embLb = nxt();
  const float *dW[3], *dal[3], *dar[3], *db[3], *dlg[3], *dlb[3];
  for (int i = 0; i < 3; ++i) {
    dW[i] = nxt(); dal[i] = nxt(); dar[i] = nxt(); db[i] = nxt();
    dlg[i] = nxt(); dlb[i] = nxt();
  }
  const float *plW1[3], *plb1[3], *plW2[3], *plb2[3];
  for (int i = 0; i < 3; ++i) { plW1[i] = nxt(); plb1[i] = nxt(); plW2[i] = nxt(); plb2[i] = nxt(); }
  const float* bW = nxt(); const float* bal = nxt(); const float* bar = nxt(); const float* bb = nxt();
  const float* up0W = nxt(); const float* up0b = nxt();
  const float *usW[2], *usb[2], *unW[2], *unb[2], *uprW[2], *uprb[2], *ulg[2], *ulb[2];
  for (int i = 0; i < 2; ++i) {
    usW[i] = nxt(); usb[i] = nxt(); unW[i] = nxt(); unb[i] = nxt();
    uprW[i] = nxt(); uprb[i] = nxt(); ulg[i] = nxt(); ulb[i] = nxt();
  }
  const float* c1W = nxt(); const float* c1b = nxt();
  const float* c1g = nxt(); const float* c1lb = nxt();
  const float* c2W = nxt(); const float* c2b = nxt();
  const float* c2g = nxt(); const float* c2lb = nxt();
  const float* c3W = nxt(); const float* c3b = nxt();
  (void)usb; (void)n_in; (void)ws_size; (void)out_size;

  // ---- pooled sizes k_i = max(1, int(0.5 * 0.8^i * n)) ----
  int cur = n0, kk[3];
  for (int i = 0; i < 3; ++i) {
    int k = (int)(0.5 * pow(0.8, (double)i) * (double)cur);
    if (k < 1) k = 1;
    kk[i] = k; cur = k;
  }
  const int k1 = kk[0], k2 = kk[1], k3 = kk[2];

  // ---- workspace ----
  char* base = (char*)d_ws; size_t off = 0;
  auto alloc = [&](size_t bytes) -> void* {
    void* p = base + off; off = (off + bytes + 255) & ~(size_t)255; return p;
  };
  float* lvlx0 = (float*)alloc((size_t)n0 * HH * 4);
  float* lvlx1 = (float*)alloc((size_t)k1 * HH * 4);
  float* lvlx2 = (float*)alloc((size_t)k2 * HH * 4);
  float* lvlx3 = (float*)alloc((size_t)k3 * HH * 4);
  float* upA = (float*)alloc((size_t)n0 * HH * 4);
  float* upB = (float*)alloc((size_t)n0 * HH * 4);
  float* buf_h = (float*)alloc((size_t)n0 * HH * 4);
  float* buf_t = (float*)alloc((size_t)n0 * HH * 4);
  float* buf_z = (float*)alloc((size_t)n0 * 256 * 4);
  float* buf_out = (float*)alloc((size_t)n0 * 256 * 4);
  float* buf_cat = (float*)alloc((size_t)n0 * 128 * 4);
  float* buf_e = (float*)alloc((size_t)E * 4 * 4);
  float* buf_el = (float*)alloc((size_t)n0 * 4 * 4);
  float* buf_er = (float*)alloc((size_t)n0 * 4 * 4);
  unsigned* buf_m = (unsigned*)alloc((size_t)n0 * 4 * 4);
  float* buf_s = (float*)alloc((size_t)n0 * 4 * 4);
  float* scores = (float*)alloc((size_t)n0 * 4);
  unsigned* encb = (unsigned*)alloc((size_t)n0 * 4);
  int* mpbuf = (int*)alloc((size_t)n0 * 4);
  float* dego = (float*)alloc((size_t)n0 * 4);
  float* degi = (float*)alloc((size_t)n0 * 4);
  int* idxL0 = (int*)alloc((size_t)k1 * 4);
  int* idxL1 = (int*)alloc((size_t)k2 * 4);
  int* idxL2 = (int*)alloc((size_t)k3 * 4);
  int* srcL1 = (int*)alloc((size_t)E * 4); int* dstL1 = (int*)alloc((size_t)E * 4);
  int* mskL1 = (int*)alloc((size_t)E * 4);
  int* srcL2 = (int*)alloc((size_t)E * 4); int* dstL2 = (int*)alloc((size_t)E * 4);
  int* mskL2 = (int*)alloc((size_t)E * 4);
  int* srcL3 = (int*)alloc((size_t)E * 4); int* dstL3 = (int*)alloc((size_t)E * 4);
  int* mskL3 = (int*)alloc((size_t)E * 4);
  unsigned* state = (unsigned*)alloc(256);

  // ---- launch helpers ----
  auto G1 = [](long long t) { return dim3((unsigned)((t + 255) / 256)); };
  auto gemm = [&](const float* X, const float* W, const float* B, float* Y,
                  int M, int K, int N, int acc) {
    dim3 g((M + 15) / 16, (N + 15) / 16);
    k_gemm<<<g, 32, 0, stream>>>(X, W, B, Y, M, K, N, acc);
  };
  auto lnact = [&](float* Y, const float* X, const float* R, const float* g,
                   const float* b, int M, int D, int act) {
    k_ln_act<<<dim3((M + 7) / 8), 256, 0, stream>>>(Y, X, R, g, b, M, D, act);
  };
  auto fillf = [&](float* p, float v, long long n) {
    k_fill_f32<<<G1(n), 256, 0, stream>>>(p, v, (int)n);
  };

  auto run_gat = [&](const float* x, int nn, const int* sp, const int* dp, const int* mp,
                     const float* W, const float* al, const float* ar, const float* bbias,
                     float* hout) {
    gemm(x, W, nullptr, buf_z, nn, HH, 256, 0);                        // z = x@W
    k_head_dots<<<G1((long long)nn * 4 * 32), 256, 0, stream>>>(buf_z, al, ar, buf_el, buf_er, nn);
    k_fill_u32<<<G1(nn * 4), 256, 0, stream>>>(buf_m, 0u, nn * 4);
    fillf(buf_s, 0.f, nn * 4);
    k_edge_logits<<<G1((long long)E * 4), 256, 0, stream>>>(buf_el, buf_er, sp, dp, mp, buf_e, buf_m, E);
    k_edge_expsum<<<G1((long long)E * 4), 256, 0, stream>>>(buf_e, buf_m, buf_s, dp, mp, E);
    fillf(buf_out, 0.f, (long long)nn * 256);
    k_gat_scatter<<<G1((long long)E * 32), 256, 0, stream>>>(buf_z, buf_e, buf_s, sp, dp, mp, buf_out, E);
    k_hmean<<<G1((long long)nn * HH), 256, 0, stream>>>(buf_out, bbias, hout, nn);
  };

  auto run_pool = [&](const float* x, int nn, int kkk, const int* sp, const int* dp,
                      const int* mp, const float* l1W, const float* l1b,
                      const float* l2W, const float* l2b, float* nx,
                      int* nsrc, int* ndst, int* nmask, int* idxout) {
    gemm(x, l1W, l1b, buf_z, nn, HH, 256, 0);
    k_leaky<<<G1((long long)nn * 256), 256, 0, stream>>>(buf_z, nn * 256);
    gemm(buf_z, l2W, l2b, buf_el, nn, 256, 4, 0);
    k_score_enc<<<G1(nn), 256, 0, stream>>>(buf_el, scores, encb, nn);
    k_bisect_init<<<1, 1, 0, stream>>>(state, (unsigned)kkk);
    for (int it = 0; it < 32; ++it) {
      k_count_ge<<<G1(nn), 256, 0, stream>>>(encb, nn, state);
      k_bisect_step<<<1, 1, 0, stream>>>(state);
    }
    k_sel_gt<<<G1(nn), 256, 0, stream>>>(encb, nn, state, idxout, kkk);
    k_sel_eq<<<G1(nn), 256, 0, stream>>>(encb, nn, state, idxout, kkk);
    k_fill_i32<<<G1(nn), 256, 0, stream>>>(mpbuf, -1, nn);
    k_build_map<<<G1(kkk), 256, 0, stream>>>(idxout, mpbuf, kkk);
    k_gather_scale<<<G1((long long)kkk * HH), 256, 0, stream>>>(x, idxout, scores, nx, kkk);
    k_remap<<<G1(E), 256, 0, stream>>>(sp, dp, mp, mpbuf, nsrc, ndst, nmask, E);
  };

  // ================= forward =================
  // embedding: gelu(ln(feat@W + b))
  gemm(feat, embW, embB, buf_h, n0, 32, HH, 0);
  lnact(lvlx0, buf_h, nullptr, embG, embLb, n0, HH, 1);

  const int* SRC[4] = { src0, srcL1, srcL2, srcL3 };
  const int* DST[4] = { dst0, dstL1, dstL2, dstL3 };
  const int* MSK[4] = { nullptr, mskL1, mskL2, mskL3 };
  int* WSRC[4] = { nullptr, srcL1, srcL2, srcL3 };
  int* WDST[4] = { nullptr, dstL1, dstL2, dstL3 };
  int* WMSK[4] = { nullptr, mskL1, mskL2, mskL3 };
  float* LX[4] = { lvlx0, lvlx1, lvlx2, lvlx3 };
  int NN[4] = { n0, k1, k2, k3 };
  int* IDX[3] = { idxL0, idxL1, idxL2 };

  // down path
  for (int i = 0; i < 3; ++i) {
    run_gat(LX[i], NN[i], SRC[i], DST[i], MSK[i], dW[i], dal[i], dar[i], db[i], buf_h);
    lnact(LX[i], buf_h, LX[i], dlg[i], dlb[i], NN[i], HH, 1);  // residual + LN + GELU
    run_pool(LX[i], NN[i], NN[i + 1], SRC[i], DST[i], MSK[i],
             plW1[i], plb1[i], plW2[i], plb2[i],
             LX[i + 1], WSRC[i + 1], WDST[i + 1], WMSK[i + 1], IDX[i]);
  }

  // bottleneck GAT (mean over heads only) -> upA[:k3]
  run_gat(lvlx3, k3, srcL3, dstL3, mskL3, bW, bal, bar, bb, upA);

  // ---- up i=0: GCN at level 2 ----
  fillf(upB, 0.f, (long long)k2 * HH);
  k_scatter_rows<<<G1((long long)k3 * HH), 256, 0, stream>>>(upA, idxL2, upB, k3);
  k_copy64<<<G1((long long)k2 * HH), 256, 0, stream>>>(buf_cat, upB, k2, 0);
  k_copy64<<<G1((long long)k2 * HH), 256, 0, stream>>>(buf_cat, lvlx2, k2, HH);
  fillf(dego, 0.f, k2); fillf(degi, 0.f, k2);
  k_degree<<<G1(E), 256, 0, stream>>>(srcL2, dstL2, mskL2, dego, degi, E);
  gemm(buf_cat, up0W, nullptr, buf_h, k2, 128, HH, 0);
  fillf(buf_t, 0.f, (long long)k2 * HH);
  k_gcn_scatter<<<G1((long long)E * 32), 256, 0, stream>>>(buf_h, srcL2, dstL2, mskL2, dego, buf_t, E);
  k_gcn_finish<<<G1((long long)k2 * HH), 256, 0, stream>>>(buf_t, degi, up0b, upA, k2);

  // ---- up i=1,2: SAGE ----
  for (int u = 0; u < 2; ++u) {
    int nj = (u == 0) ? k1 : n0;
    int km = (u == 0) ? k2 : k1;
    const int* idp = (u == 0) ? idxL1 : idxL0;
    const float* xs = (u == 0) ? lvlx1 : lvlx0;
    const int* es = (u == 0) ? srcL1 : src0;
    const int* ed = (u == 0) ? dstL1 : dst0;
    const int* em = (u == 0) ? mskL1 : nullptr;
    fillf(upB, 0.f, (long long)nj * HH);
    k_scatter_rows<<<G1((long long)km * HH), 256, 0, stream>>>(upA, idp, upB, km);
    k_copy64<<<G1((long long)nj * HH), 256, 0, stream>>>(buf_cat, upB, nj, 0);
    k_copy64<<<G1((long long)nj * HH), 256, 0, stream>>>(buf_cat, xs, nj, HH);
    fillf(buf_out, 0.f, (long long)nj * 128);
    fillf(dego, 0.f, nj);
    k_sage_scatter<<<G1((long long)E * 32), 256, 0, stream>>>(buf_cat, es, ed, em, buf_out, dego, E);
    k_sage_div<<<G1((long long)nj * 128), 256, 0, stream>>>(buf_out, dego, buf_z, nj);
    gemm(buf_cat, usW[u], nullptr, buf_h, nj, 128, HH, 0);   // cat @ self.W
    gemm(buf_z, unW[u], unb[u], buf_h, nj, 128, HH, 1);      // += mean @ neigh.W + neigh.b
    gemm(buf_cat, uprW[u], uprb[u], buf_t, nj, 128, HH, 0);  // res = cat @ proj.W + proj.b
    lnact(upA, buf_h, buf_t, ulg[u], ulb[u], nj, HH, 1);     // gelu(ln(h + res))
  }

  // ---- classifier ----
  gemm(upA, c1W, c1b, buf_cat, n0, HH, 128, 0);
  lnact(buf_cat, buf_cat, nullptr, c1g, c1lb, n0, 128, 1);
  gemm(buf_cat, c2W, c2b, upB, n0, 128, HH, 0);
  lnact(upB, upB, nullptr, c2g, c2lb, n0, HH, 1);
  gemm(upB, c3W, c3b, (float*)d_out, n0, HH, 2, 0);
}